// DE_89404039234069
// MI455X (gfx1250) — compile-verified
//
#include <hip/hip_runtime.h>
#include <hip/hip_bf16.h>

// MI455X / gfx1250, wave32. Nearest-neighbor derivative + batchnorm.
// Kernel 1: one wave per 16-row block; exact (xi - xj) difference tiles on the
//           matrix unit via V_WMMA_F32_16X16X4_F32; VALU running argmin.
//           Validity masking hoisted to the single "diagonal" tile; B-operand
//           LDS reads software-pipelined one tile ahead; wave-uniform loop
//           bounds forced scalar via readfirstlane.
// Kernel 2: single-block mean/var reduction + normalization.

typedef __attribute__((ext_vector_type(2))) float v2f;
typedef __attribute__((ext_vector_type(8))) float v8f;

#if defined(__has_builtin)
#  if __has_builtin(__builtin_amdgcn_wmma_f32_16x16x4_f32)
#    define HAVE_WMMA_F32_4 1
#  endif
#endif
#ifndef HAVE_WMMA_F32_4
#  define HAVE_WMMA_F32_4 0
#endif

#define NBATCH   8
#define NPTS     2048
#define NC       1024
#define NTOT     (NBATCH * NPTS)      // 16384 rows total
#define EPS_D    2e-6f
#define BN_EPS   1e-5f
#define BIGF     3.0e38f

// Output layout (flat fp32, return-order concat):
//   y_diff [0, 16384)  x_diff [16384, 32768)  d_out (B,2048,2) [32768, 65536)
//   x_n [65536, 81920)  y_n [81920, 98304)
#define OFF_YD   0
#define OFF_XD   16384
#define OFF_D    32768
#define OFF_XN   65536
#define OFF_YN   81920

__global__ __launch_bounds__(256) void nn_deriv(const float* __restrict__ yg,
                                                const float* __restrict__ xg,
                                                float* __restrict__ out,
                                                float* __restrict__ d2ws)
{
    __shared__ float xs[NPTS];
    __shared__ int   residx[8][16];

    const int tid = threadIdx.x;
    const int b   = blockIdx.x >> 4;            // 16 HIP blocks per batch
    for (int i = tid; i < NPTS; i += 256) xs[i] = xg[b * NPTS + i];
    __syncthreads();

    const int  lane = tid & 31;
    const int  w    = tid >> 5;                 // wave in block (0..7)
    const int  lc   = lane & 15;
    const bool hi   = lane >= 16;
    const int  m0   = ((blockIdx.x & 15) * 8 + w) * 16;   // first row of this wave's block
    // Wave-uniform scan parameters -> SGPRs, so tile loops are scalar loops.
    const int  diagU = __builtin_amdgcn_readfirstlane(m0 >> 4); // only masked tile
    const int  isTU  = __builtin_amdgcn_readfirstlane(m0 >= NC ? 1 : 0);

    int mrow[8];                                // row index held by (this lane, state v)
    #pragma unroll
    for (int v = 0; v < 8; ++v) mrow[v] = m0 + v + (hi ? 8 : 0);

    float minv[8];
    int   mini[8];
    #pragma unroll
    for (int v = 0; v < 8; ++v) { minv[v] = BIGF; mini[v] = 0; }

#if HAVE_WMMA_F32_4
    // A (16x4 f32): lo-half VGPR0 = K0 row = x_i, lo-half VGPR1 = K1 row = -1.
    // hi-half slots are K2/K3 rows: MUST be zero (B's K2 row is all-ones below).
    v2f A;
    A[0] = hi ? 0.0f : xs[m0 + lc];
    A[1] = hi ? 0.0f : -1.0f;
#else
    float xrow[8];
    #pragma unroll
    for (int v = 0; v < 8; ++v) xrow[v] = xs[mrow[v]];
#endif

    // One tile: columns [c0, c0+16). `xb` = xs[c0 + lc], loaded one tile ahead.
    auto tile_unmasked = [&](int c0, float xb) {
        const int c = c0 + lc;
#if HAVE_WMMA_F32_4
        v2f Bv;
        Bv[0] = 1.0f;      // K0 row = 1 (hi lanes feed K2, killed by A's zeros)
        Bv[1] = xb;        // K1 row = x_j (hi lanes feed K3, killed by A's zeros)
        v8f D = {0.f, 0.f, 0.f, 0.f, 0.f, 0.f, 0.f, 0.f};
        // D[i][j] = x_i*1 + (-1)*x_j = fl(x_i - x_j)  (exact products, one rounding)
        D = __builtin_amdgcn_wmma_f32_16x16x4_f32(false, A, false, Bv,
                                                  (short)0, D, false, false);
#endif
        #pragma unroll
        for (int v = 0; v < 8; ++v) {
#if HAVE_WMMA_F32_4
            const float dj = fabsf(D[v]);
#else
            const float dj = fabsf(xrow[v] - xb);
#endif
            const bool better = dj < minv[v];   // strict < : ties keep lowest c
            minv[v] = better ? dj : minv[v];
            mini[v] = better ? c  : mini[v];
        }
    };

    auto tile_masked = [&](int c0, float xb) {
        const int c = c0 + lc;
#if HAVE_WMMA_F32_4
        v2f Bv;
        Bv[0] = 1.0f;
        Bv[1] = xb;
        v8f D = {0.f, 0.f, 0.f, 0.f, 0.f, 0.f, 0.f, 0.f};
        D = __builtin_amdgcn_wmma_f32_16x16x4_f32(false, A, false, Bv,
                                                  (short)0, D, false, false);
#endif
        #pragma unroll
        for (int v = 0; v < 8; ++v) {
#if HAVE_WMMA_F32_4
            float dj = fabsf(D[v]);
#else
            float dj = fabsf(xrow[v] - xb);
#endif
            // context: exclude self (c != m); target: causal cut (c < m)
            const bool valid = isTU ? (c < mrow[v]) : (c != mrow[v]);
            dj = valid ? dj : BIGF;
            const bool better = dj < minv[v];
            minv[v] = better ? dj : minv[v];
            mini[v] = better ? c  : mini[v];
        }
    };

    // Segment 1: tiles [0, diagU) — every column valid for every row in block.
    // Pipelined: load tile t+1's B value while updating tile t.
    float xb = xs[lc];                           // tile 0 (valid even if diagU==0)
    for (int t = 0; t < diagU; ++t) {
        const float cur = xb;
        xb = xs[((t + 1) << 4) + lc];            // t+1 <= diagU <= 127 -> idx <= 2047
        tile_unmasked(t << 4, cur);
    }
    // Segment 2: the diagonal tile (xb already holds its data).
    tile_masked(diagU << 4, xb);
    // Segment 3 (context blocks only): tiles (diagU, 64).
    if (!isTU) {
        float xb2 = xs[((diagU + 1) << 4) + lc]; // idx <= 1024+15, in-bounds
        for (int t = diagU + 1; t < NC / 16; ++t) {
            const float cur = xb2;
            xb2 = xs[((t + 1) << 4) + lc];       // t+1 <= 64 -> idx <= 1039, harmless
            tile_unmasked(t << 4, cur);
        }
    }

    // Lexicographic (dist, idx) argmin butterfly within each 16-lane half.
    #pragma unroll
    for (int v = 0; v < 8; ++v) {
        #pragma unroll
        for (int s = 8; s >= 1; s >>= 1) {
            const float ov = __shfl_xor(minv[v], s, 32);
            const int   oi = __shfl_xor(mini[v], s, 32);
            if (ov < minv[v] || (ov == minv[v] && oi < mini[v])) {
                minv[v] = ov; mini[v] = oi;
            }
        }
    }

    if (lane == 0 || lane == 16) {
        const int base = hi ? 8 : 0;
        #pragma unroll
        for (int v = 0; v < 8; ++v) residx[w][base + v] = mini[v];
    }
    __syncthreads();

    if (lane < 16) {
        const int r  = m0 + lane;               // row in batch
        const int jn = residx[w][lane];         // nearest-neighbor index
        const float xr = xs[r],  xn = xs[jn];
        const float yr = yg[b * NPTS + r], yn = yg[b * NPTS + jn];
        const float xd = xr - xn, yd = yr - yn;
        const float d  = yd / (EPS_D + fabsf(xd));   // sqrtf(x*x) == fabsf(x) in RN
        const float d1 = (d != d) ? 10000.0f : d;
        const float d2 = (fabsf(d) > 200.0f) ? 0.0f : d;   // NaN passes through here
        const float label = (d2 == d1) ? 1.0f : 0.0f;      // NaN compares false -> 0
        const int g = b * NPTS + r;
        out[OFF_YD + g]         = yd;
        out[OFF_XD + g]         = xd;
        out[OFF_D  + 2 * g + 1] = label;
        out[OFF_XN + g]         = xn;
        out[OFF_YN + g]         = yn;
        d2ws[g] = d2;
    }
}

__global__ __launch_bounds__(1024) void bn_norm(const float* __restrict__ d2ws,
                                                const float* __restrict__ wgt,
                                                const float* __restrict__ bias,
                                                float* __restrict__ out)
{
    __shared__ float red[32];
    const int tid = threadIdx.x;

    float v[16];
    float s = 0.0f;
    #pragma unroll
    for (int k = 0; k < 16; ++k) { v[k] = d2ws[tid + 1024 * k]; s += v[k]; }
    #pragma unroll
    for (int o = 16; o >= 1; o >>= 1) s += __shfl_xor(s, o, 32);
    if ((tid & 31) == 0) red[tid >> 5] = s;
    __syncthreads();
    if (tid < 32) {
        float t2 = red[tid];
        #pragma unroll
        for (int o = 16; o >= 1; o >>= 1) t2 += __shfl_xor(t2, o, 32);
        if (tid == 0) red[0] = t2;
    }
    __syncthreads();
    const float mean = red[0] * (1.0f / (float)NTOT);
    __syncthreads();

    float s2 = 0.0f;
    #pragma unroll
    for (int k = 0; k < 16; ++k) { const float t = v[k] - mean; s2 += t * t; }
    #pragma unroll
    for (int o = 16; o >= 1; o >>= 1) s2 += __shfl_xor(s2, o, 32);
    if ((tid & 31) == 0) red[tid >> 5] = s2;
    __syncthreads();
    if (tid < 32) {
        float t2 = red[tid];
        #pragma unroll
        for (int o = 16; o >= 1; o >>= 1) t2 += __shfl_xor(t2, o, 32);
        if (tid == 0) red[0] = t2;
    }
    __syncthreads();
    const float var    = red[0] * (1.0f / (float)NTOT);
    const float invstd = 1.0f / sqrtf(var + BN_EPS);
    const float w0 = wgt[0], b0 = bias[0];
    #pragma unroll
    for (int k = 0; k < 16; ++k) {
        const int g = tid + 1024 * k;
        out[OFF_D + 2 * g] = (v[k] - mean) * invstd * w0 + b0;
    }
}

extern "C" void kernel_launch(void* const* d_in, const int* in_sizes, int n_in,
                              void* d_out, int out_size, void* d_ws, size_t ws_size,
                              hipStream_t stream)
{
    const float* y    = (const float*)d_in[0];
    const float* x    = (const float*)d_in[1];
    const float* wgt  = (const float*)d_in[2];
    const float* bias = (const float*)d_in[3];
    // d_in[4]=n_C, d_in[5]=n_T are compile-time constants (1024, 1024) here.
    float* out = (float*)d_out;
    float* ws  = (float*)d_ws;    // 16384 floats of scratch for clamped derivative

    nn_deriv<<<128, 256, 0, stream>>>(y, x, out, ws);
    bn_norm<<<1, 1024, 0, stream>>>(ws, wgt, bias, out);

    (void)in_sizes; (void)n_in; (void)out_size; (void)ws_size;
}